// Net_27513560498431
// MI455X (gfx1250) — compile-verified
//
#include <hip/hip_runtime.h>
#include <math.h>

typedef __attribute__((ext_vector_type(2))) float v2f;
typedef __attribute__((ext_vector_type(8))) float v8f;

#define F_IN 300
#define HDIM 128
#define CDIM 2

// ---------------------------------------------------------------- utilities
__global__ __launch_bounds__(256) void zero_f32(float* __restrict__ p, size_t n) {
    size_t i = (size_t)blockIdx.x * blockDim.x + threadIdx.x;
    if (i < n) p[i] = 0.0f;
}

// deg[dst] += 1 per edge
__global__ __launch_bounds__(256) void deg_kernel(const int* __restrict__ ei, int E,
                                                  float* __restrict__ deg) {
    int e = blockIdx.x * blockDim.x + threadIdx.x;
    if (e < E) atomicAdd(&deg[ei[E + e]], 1.0f);
}

// root[g] = first node of graph g (batch sorted, all graphs present)
__global__ __launch_bounds__(256) void root_kernel(const int* __restrict__ batch, int N,
                                                   int* __restrict__ root) {
    int i = blockIdx.x * blockDim.x + threadIdx.x;
    if (i < N) {
        int g = batch[i];
        if (i == 0 || batch[i - 1] != g) root[g] = i;
    }
}

// ------------------------------------------------- fused GEMM: xlr = x @ [Wl^T | Wr^T]
// C is [N, 256]: cols 0..127 = x@Wl^T, cols 128..255 = x@Wr^T.
// Workgroup = 256 threads = 8 waves. M-tile = 64 nodes (4 x 16 sub-tiles),
// each wave owns two 16-col tiles -> 8 v8f accumulators (64 VGPRs).
// One B-fragment load feeds 4 WMMAs (4x less L2 traffic on W vs M-tile=16).
// K=300 swept 4 at a time with V_WMMA_F32_16X16X4_F32.
__global__ __launch_bounds__(256) void gemm_xlr(const float* __restrict__ x,
                                                const float* __restrict__ Wl,
                                                const float* __restrict__ Wr,
                                                float* __restrict__ xlr, int N) {
    constexpr int AS = 308;               // padded LDS row stride; 308*4 B = 77*16 B
    __shared__ float Ax[64 * AS];         // 64 x 300 A tile, 77 KB

    const int m0 = blockIdx.x * 64;
    // float4 staging: each row is exactly 75 float4s (300/4), stride keeps 16B align
    for (int t = threadIdx.x; t < 64 * 75; t += 256) {
        int r = t / 75, c4 = (t - r * 75) * 4;
        int row = m0 + r; if (row >= N) row = N - 1;   // clamp (tail block only)
        const float4 v = *(const float4*)(x + (size_t)row * F_IN + c4);
        *(float4*)(&Ax[r * AS + c4]) = v;
    }
    __syncthreads();

    const int wave = threadIdx.x >> 5;
    const int lane = threadIdx.x & 31;
    const int mrow = lane & 15;           // M (A) / N (B) index within a tile
    const int kh   = lane >> 4;           // K half-select

    const int c0 = wave * 32;             // tile-0 columns
    const int c1 = c0 + 16;               // tile-1 columns
    const float* Wrow0 = ((c0 < HDIM) ? (Wl + (size_t)c0 * F_IN)
                                      : (Wr + (size_t)(c0 - HDIM) * F_IN))
                         + (size_t)mrow * F_IN;
    const float* Wrow1 = ((c1 < HDIM) ? (Wl + (size_t)c1 * F_IN)
                                      : (Wr + (size_t)(c1 - HDIM) * F_IN))
                         + (size_t)mrow * F_IN;

    v8f acc[4][2] = {};
    for (int k0 = 0; k0 < F_IN; k0 += 4) {
        const int k = k0 + 2 * kh;
        v2f b0; b0.x = Wrow0[k]; b0.y = Wrow0[k + 1];
        v2f b1; b1.x = Wrow1[k]; b1.y = Wrow1[k + 1];
        #pragma unroll
        for (int ms = 0; ms < 4; ++ms) {
            const float* Ar = &Ax[(ms * 16 + mrow) * AS];
            v2f a; a.x = Ar[k]; a.y = Ar[k + 1];
            acc[ms][0] = __builtin_amdgcn_wmma_f32_16x16x4_f32(
                false, a, false, b0, (short)0, acc[ms][0], false, false);
            acc[ms][1] = __builtin_amdgcn_wmma_f32_16x16x4_f32(
                false, a, false, b1, (short)0, acc[ms][1], false, false);
        }
    }

    // C layout: VGPR v -> row (v + 8*kh), col = lane&15
    if (m0 + 64 <= N) {                   // common path: whole tile in range
        #pragma unroll
        for (int ms = 0; ms < 4; ++ms) {
            #pragma unroll
            for (int v = 0; v < 8; ++v) {
                int m = m0 + ms * 16 + v + 8 * kh;
                xlr[(size_t)m * 256 + c0 + mrow] = acc[ms][0][v];
                xlr[(size_t)m * 256 + c1 + mrow] = acc[ms][1][v];
            }
        }
    } else {                              // single remainder block
        for (int ms = 0; ms < 4; ++ms) {
            for (int v = 0; v < 8; ++v) {
                int m = m0 + ms * 16 + v + 8 * kh;
                if (m < N) {
                    xlr[(size_t)m * 256 + c0 + mrow] = acc[ms][0][v];
                    xlr[(size_t)m * 256 + c1 + mrow] = acc[ms][1][v];
                }
            }
        }
    }
}

// ------------------------------------------------- edge scatter in H-space
// hsum[dst, :] += (x@Wl^T)[src, :]   (two edges per 256-thread block)
__global__ __launch_bounds__(256) void scatter_kernel(const int* __restrict__ ei, int E,
                                                      const float* __restrict__ xlr,
                                                      float* __restrict__ hsum) {
    int e = blockIdx.x * 2 + (threadIdx.x >> 7);
    int j = threadIdx.x & 127;
    if (e < E) {
        int s = ei[e];
        int d = ei[E + e];
        atomicAdd(&hsum[(size_t)d * HDIM + j], xlr[(size_t)s * 256 + j]);
    }
}

// ------------------------------------------------- combine + segment-max pool
// h[n,j] = relu(hsum[n,j]/max(deg,1) + bl[j] + (x@Wr^T)[n,j]); hg[g,j]=max over graph.
// batch is sorted -> running max per thread, one atomic per segment boundary.
// relu >= 0 and hg zero-initialized, so uint-bitpattern atomicMax is exact.
__global__ __launch_bounds__(128) void combine_pool(const float* __restrict__ hsum,
                                                    const float* __restrict__ xlr,
                                                    const float* __restrict__ deg,
                                                    const float* __restrict__ bl,
                                                    const int* __restrict__ batch,
                                                    unsigned* __restrict__ hg_bits,
                                                    int N) {
    const int j = threadIdx.x;
    const int base = blockIdx.x * 128;
    const float blj = bl[j];
    int curg = -1;
    float m = 0.0f;
    for (int i = 0; i < 128; ++i) {
        int n = base + i;
        if (n >= N) break;
        int g = batch[n];
        if (g != curg) {
            if (curg >= 0)
                atomicMax(&hg_bits[(size_t)curg * HDIM + j], __float_as_uint(m));
            curg = g;
            m = 0.0f;
        }
        float d = deg[n];
        float v = hsum[(size_t)n * HDIM + j] / fmaxf(d, 1.0f)
                + xlr[(size_t)n * 256 + HDIM + j] + blj;
        v = fmaxf(v, 0.0f);
        m = fmaxf(m, v);
    }
    if (curg >= 0)
        atomicMax(&hg_bits[(size_t)curg * HDIM + j], __float_as_uint(m));
}

// ------------------------------------------------- MLP head, one block per graph
__global__ __launch_bounds__(128) void head_kernel(const float* __restrict__ x,
                                                   const int* __restrict__ root,
                                                   const float* __restrict__ hg,
                                                   const float* __restrict__ W0,
                                                   const float* __restrict__ b0,
                                                   const float* __restrict__ W1,
                                                   const float* __restrict__ b1,
                                                   const float* __restrict__ W2,
                                                   const float* __restrict__ b2,
                                                   float* __restrict__ out) {
    __shared__ float xr_s[F_IN];
    __shared__ float hg_s[HDIM];
    __shared__ float ns[HDIM];
    __shared__ float h1[HDIM];
    __shared__ float p0[HDIM];
    __shared__ float p1[HDIM];

    const int g = blockIdx.x;
    const int j = threadIdx.x;
    const int r = root[g];

    for (int k = j; k < F_IN; k += 128) xr_s[k] = x[(size_t)r * F_IN + k];
    hg_s[j] = hg[(size_t)g * HDIM + j];
    __syncthreads();

    // news = relu(x[root] @ W0^T + b0)
    float a0 = b0[j];
    const float* w0r = W0 + (size_t)j * F_IN;
    for (int k = 0; k < F_IN; ++k) a0 += xr_s[k] * w0r[k];
    ns[j] = fmaxf(a0, 0.0f);
    __syncthreads();

    // h1 = relu([news, hg] @ W1^T + b1)
    float a1 = b1[j];
    const float* w1r = W1 + (size_t)j * (2 * HDIM);
    for (int k = 0; k < HDIM; ++k)
        a1 += ns[k] * w1r[k] + hg_s[k] * w1r[HDIM + k];
    h1[j] = fmaxf(a1, 0.0f);
    __syncthreads();

    // logits + log_softmax (C=2)
    p0[j] = h1[j] * W2[j];
    p1[j] = h1[j] * W2[HDIM + j];
    __syncthreads();
    for (int s = 64; s > 0; s >>= 1) {
        if (j < s) { p0[j] += p0[j + s]; p1[j] += p1[j + s]; }
        __syncthreads();
    }
    if (j == 0) {
        float l0 = p0[0] + b2[0];
        float l1 = p1[0] + b2[1];
        float mx = fmaxf(l0, l1);
        float lse = mx + logf(expf(l0 - mx) + expf(l1 - mx));
        out[(size_t)g * CDIM + 0] = l0 - lse;
        out[(size_t)g * CDIM + 1] = l1 - lse;
    }
}

// ---------------------------------------------------------------- launch
extern "C" void kernel_launch(void* const* d_in, const int* in_sizes, int n_in,
                              void* d_out, int out_size, void* d_ws, size_t ws_size,
                              hipStream_t stream) {
    const float* x     = (const float*)d_in[0];
    const int*   ei    = (const int*)d_in[1];     // [2, E]
    const int*   batch = (const int*)d_in[2];     // [N]
    const float* Wl    = (const float*)d_in[3];
    const float* bl    = (const float*)d_in[4];
    const float* Wr    = (const float*)d_in[5];
    const float* W0    = (const float*)d_in[6];
    const float* b0    = (const float*)d_in[7];
    const float* W1    = (const float*)d_in[8];
    const float* b1    = (const float*)d_in[9];
    const float* W2    = (const float*)d_in[10];
    const float* b2    = (const float*)d_in[11];
    float* out = (float*)d_out;

    const int N = in_sizes[0] / F_IN;
    const int E = in_sizes[1] / 2;
    const int G = out_size / CDIM;

    // workspace layout (floats)
    float* deg  = (float*)d_ws;                          // N
    float* xlr  = deg + N;                               // N * 256
    float* hsum = xlr + (size_t)N * 256;                 // N * 128
    float* hg   = hsum + (size_t)N * HDIM;               // G * 128
    int*   root = (int*)(hg + (size_t)G * HDIM);         // G

    auto blocks = [](size_t n, int b) { return (unsigned)((n + b - 1) / b); };

    // zero-init (graph-replay safe: done every call)
    zero_f32<<<blocks((size_t)N, 256), 256, 0, stream>>>(deg, (size_t)N);
    zero_f32<<<blocks((size_t)N * HDIM, 256), 256, 0, stream>>>(hsum, (size_t)N * HDIM);
    zero_f32<<<blocks((size_t)G * HDIM, 256), 256, 0, stream>>>(hg, (size_t)G * HDIM);

    deg_kernel<<<blocks(E, 256), 256, 0, stream>>>(ei, E, deg);
    root_kernel<<<blocks(N, 256), 256, 0, stream>>>(batch, N, root);

    gemm_xlr<<<blocks(N, 64), 256, 0, stream>>>(x, Wl, Wr, xlr, N);

    scatter_kernel<<<blocks(E, 2), 256, 0, stream>>>(ei, E, xlr, hsum);

    combine_pool<<<blocks(N, 128), 128, 0, stream>>>(hsum, xlr, deg, bl, batch,
                                                     (unsigned*)hg, N);

    head_kernel<<<G, 128, 0, stream>>>(x, root, hg, W0, b0, W1, b1, W2, b2, out);
}